// RejectionSampler_18889266168367
// MI455X (gfx1250) — compile-verified
//
#include <hip/hip_runtime.h>
#include <hip/hip_bf16.h>

typedef float v4f __attribute__((ext_vector_type(4)));

#define AM_BLOCK 256   // 8 waves of 32 (wave32 on gfx1250)

// ---------------------------------------------------------------------------
// Kernel 1: row-wise argmax over [rows, vocab] fp32, first-occurrence ties.
// Bandwidth-bound: 1152 x 128000 x 4B ~= 590 MB stream -> ~25 us @ 23.3 TB/s.
// global_load_b128 th:TH_LOAD_NT (no L2 pollution of the one-shot stream),
// wave-uniform global_prefetch_b8 16 iterations ahead, wave32 ds reductions.
// Steady-state VALU ~1.2 ops/element (max3 + cmp; update block rarely taken).
// ---------------------------------------------------------------------------
__global__ __launch_bounds__(AM_BLOCK) void argmax_rows_kernel(
    const float* __restrict__ logits, int* __restrict__ am, int vocab)
{
    const int row = blockIdx.x;
    const float* __restrict__ p = logits + (size_t)row * (size_t)vocab;
    const int tid = threadIdx.x;

    float bestv = -__builtin_inff();
    int   besti = 0;

    const int CHUNK = AM_BLOCK * 4;        // 1024 floats / 4 KB per block-iter
    const int iters = vocab / CHUNK;       // 125 for vocab=128000

    for (int it = 0; it < iters; ++it) {
        const int base = it * CHUNK + tid * 4;

        // Wave-uniform prefetch: every 8th iteration all 256 lanes prefetch
        // distinct 128B lines (tid*32 floats) => 32KB = the next 8 iterations
        // of this block's stream, 16 iterations (64KB) ahead. Scalar branch,
        // no EXEC churn; guard keeps addresses inside this row.
        if (((it & 7) == 0) && (it + 24 <= iters))
            __builtin_prefetch(p + (it + 16) * CHUNK + tid * 32, 0, 0);

        // Streaming read: TH=NT so the 590MB one-shot stream bypasses L2 reuse.
        v4f v = __builtin_nontemporal_load((const v4f*)(p + base));

        // Quad max first (2x v_max3-class ops), then a single compare. The
        // update block is divergent but almost never taken after warm-up
        // (P(update) ~ 4/elements_seen per lane).
        float m01 = fmaxf(v[0], v[1]);
        float m23 = fmaxf(v[2], v[3]);
        float m   = fmaxf(m01, m23);
        if (m > bestv) {   // strict '>': earlier iteration wins exact ties
            bestv = m;
            // first-occurrence index within the quad
            besti = (v[0] == m) ? (base + 0)
                  : (v[1] == m) ? (base + 1)
                  : (v[2] == m) ? (base + 2)
                  :               (base + 3);
        }
    }
    // Generic tail (vocab % 1024 != 0); no-op for 128000.
    for (int i = iters * CHUNK + tid; i < vocab; i += AM_BLOCK) {
        float x = p[i];
        if (x > bestv) { bestv = x; besti = i; }
    }

    // wave32 butterfly reduction; on equal values prefer the smaller index
    // (reproduces jnp.argmax first-occurrence semantics globally).
    #pragma unroll
    for (int off = 16; off > 0; off >>= 1) {
        float ov = __shfl_xor(bestv, off, 32);
        int   oi = __shfl_xor(besti, off, 32);
        if (ov > bestv || (ov == bestv && oi < besti)) { bestv = ov; besti = oi; }
    }

    __shared__ float sv[AM_BLOCK / 32];
    __shared__ int   si[AM_BLOCK / 32];
    const int lane = tid & 31;
    const int wv   = tid >> 5;
    if (lane == 0) { sv[wv] = bestv; si[wv] = besti; }
    __syncthreads();
    if (tid == 0) {
        float bv = sv[0]; int bi = si[0];
        #pragma unroll
        for (int w = 1; w < AM_BLOCK / 32; ++w) {
            if (sv[w] > bv || (sv[w] == bv && si[w] < bi)) { bv = sv[w]; bi = si[w]; }
        }
        am[row] = bi;
    }
}

// ---------------------------------------------------------------------------
// Kernel 2: greedy rejection sampling over ragged sequences. One thread per
// sequence; O(B) prefix sum per thread is ~256 cached loads (negligible vs
// the 25us argmax stage). Outputs written as fp32 (token ids < 2^24 exact).
// ---------------------------------------------------------------------------
__global__ void rejection_kernel(
    const int* __restrict__ am, const int* __restrict__ draft,
    const int* __restrict__ bonus, const int* __restrict__ ndraft,
    float* __restrict__ out, int B, int S, int T)
{
    int b = blockIdx.x * blockDim.x + threadIdx.x;
    if (b >= B) return;

    int n  = ndraft[b];
    int cu = 0;
    for (int j = 0; j < b; ++j) cu += ndraft[j];

    // Leading-match count (num_accept).
    int num_accept = 0;
    for (int i = 0; i < n; ++i) {
        int t = cu + i; if (t >= T) t = T - 1;
        if (am[t] == draft[t]) ++num_accept;
        else break;
    }
    const bool rejected  = (num_accept < n);
    const int  write_len = rejected ? (num_accept + 1) : n;

    float* out_tok  = out;                         // [B, S+1]
    float* out_nrej = out + (size_t)B * (S + 1);   // [B]
    float* out_last = out_nrej + B;                // [B]

    const int bon = bonus[b];
    for (int ppos = 0; ppos <= S; ++ppos) {
        float val = -1.0f;
        if (ppos < write_len) {
            int t = cu + ppos; if (t >= T) t = T - 1;
            val = (float)am[t];                    // target argmax incl. mismatch
        } else if (ppos == n && !rejected) {
            val = (float)bon;                      // bonus on full acceptance
        }
        out_tok[(size_t)b * (S + 1) + ppos] = val;
    }

    out_nrej[b] = (float)(n - num_accept);

    if (rejected) {
        int fmp = num_accept;
        if (fmp > S - 1) fmp = S - 1;
        if (fmp < 0)     fmp = 0;
        int t = cu + fmp; if (t >= T) t = T - 1;
        out_last[b] = (float)am[t];
    } else {
        out_last[b] = (float)bon;
    }
}

// ---------------------------------------------------------------------------
extern "C" void kernel_launch(void* const* d_in, const int* in_sizes, int n_in,
                              void* d_out, int out_size, void* d_ws, size_t ws_size,
                              hipStream_t stream)
{
    const float* logits = (const float*)d_in[0];   // [T, V] fp32
    const int*   draft  = (const int*)  d_in[1];   // [T]
    const int*   bonus  = (const int*)  d_in[2];   // [B]
    const int*   ndraft = (const int*)  d_in[3];   // [B]
    // d_in[4] = max_spec_num scalar (device mem) -- recovered from out_size
    // instead so we stay graph-capture safe (no sync reads).

    const int T = in_sizes[1];                 // num_tokens (draft ids length)
    const int V = in_sizes[0] / T;             // vocab
    const int B = in_sizes[2];                 // batch (bonus ids length)
    const int S = out_size / B - 3;            // out_size = B*(S+1) + 2B

    int* am = (int*)d_ws;                      // per-row argmax (T ints)

    argmax_rows_kernel<<<T, AM_BLOCK, 0, stream>>>(logits, am, V);
    rejection_kernel<<<(B + 255) / 256, 256, 0, stream>>>(
        am, draft, bonus, ndraft, (float*)d_out, B, S, T);
}